// causal_score_45174466020067
// MI455X (gfx1250) — compile-verified
//
#include <hip/hip_runtime.h>
#include <hip/hip_bf16.h>

// ---------------------------------------------------------------------------
// MI455X (gfx1250, wave32) implementation.
//
// Key algebra: noise cancels in (p_mask - p_orig); only T=10 columns of
// W_model are needed. Everything reduces to two skinny fp32 GEMMs over the
// masks tensor, both mapped to V_WMMA_F32_16X16X4_F32 (exact fp32 — the
// problem is HBM-bound at ~0.5 GB total traffic, so fp32 WMMA is free).
// ---------------------------------------------------------------------------

#define NN 500
#define HH 224
#define WW 224
#define HWSZ (HH * WW)        // 50176
#define TT 10
#define CC 980
#define NPAD 512
#define KCHUNKS 32            // K-split for GEMM1 (HWSZ/32 = 1568, %4 == 0)
#define LDPAD 516             // LDS row stride for p_final (kills bank conflicts)

typedef float v2f __attribute__((ext_vector_type(2)));
typedef float v8f __attribute__((ext_vector_type(8)));

// ---------------------------------------------------------------------------
// ws layout (floats):
//   Apad [16][HWSZ]      : A[t,hw], rows 10..15 zero
//   dcol [HWSZ]          : sum_n masks + 1e-6
//   S    [16]            : sum_hw A[t,hw]
//   diff [NPAD][16]      : GEMM1 output (atomic accumulated)
//   PF   [16][NPAD]      : p_final padded, t-major
// ---------------------------------------------------------------------------
#define OFF_APAD 0
#define OFF_DCOL (OFF_APAD + 16 * HWSZ)
#define OFF_S    (OFF_DCOL + HWSZ)
#define OFF_DIFF (OFF_S + 16)
#define OFF_PF   (OFF_DIFF + NPAD * 16)

__global__ void init_zero_kernel(float* diff, float* S) {
    int i = blockIdx.x * blockDim.x + threadIdx.x;
    if (i < NPAD * 16) diff[i] = 0.0f;
    if (i < 16) S[i] = 0.0f;
}

// A[t,hw] = sum_c x0[c,hw] * W[(c*HW+hw)*C + tc[t]]  ; also S[t] = sum_hw A
__global__ void build_A_kernel(const float* __restrict__ x0,
                               const float* __restrict__ Wm,
                               const int* __restrict__ tc,
                               float* __restrict__ Apad,
                               float* __restrict__ S) {
    __shared__ float ssum[16];
    if (threadIdx.x < 16) ssum[threadIdx.x] = 0.0f;
    __syncthreads();
    int hw = blockIdx.x * blockDim.x + threadIdx.x;
    if (hw < HWSZ) {
        float x0v = x0[hw];
        float x1v = x0[HWSZ + hw];
        float x2v = x0[2 * HWSZ + hw];
        size_t r0 = (size_t)hw * CC;
        size_t r1 = ((size_t)HWSZ + hw) * CC;
        size_t r2 = ((size_t)2 * HWSZ + hw) * CC;
        for (int t = 0; t < 16; ++t) {
            float a = 0.0f;
            if (t < TT) {
                int col = tc[t];
                col = col < 0 ? 0 : (col >= CC ? CC - 1 : col);
                a = x0v * Wm[r0 + col] + x1v * Wm[r1 + col] + x2v * Wm[r2 + col];
                atomicAdd(&ssum[t], a);
            }
            Apad[(size_t)t * HWSZ + hw] = a;
        }
    }
    __syncthreads();
    if (threadIdx.x < TT) atomicAdd(&S[threadIdx.x], ssum[threadIdx.x]);
}

// dcol[hw] = sum_n masks[n,hw] + 1e-6   (fully coalesced column walk)
__global__ void colsum_kernel(const float* __restrict__ masks,
                              float* __restrict__ dcol) {
    int hw = blockIdx.x * blockDim.x + threadIdx.x;
    if (hw >= HWSZ) return;
    float s = 0.0f;
    for (int n = 0; n < NN; ++n) s += masks[(size_t)n * HWSZ + hw];
    dcol[hw] = s + 1e-6f;
}

// GEMM1: diff[n,t] += sum_{hw in chunk} masks[n,hw] * Apad[t,hw]
// One wave per (n-block of 16) x (K chunk). V_WMMA_F32_16X16X4_F32.
__global__ void __launch_bounds__(32)
gemm_diff_kernel(const float* __restrict__ masks,
                 const float* __restrict__ Apad,
                 float* __restrict__ diff) {
    int nblk = blockIdx.x;            // 0..31
    int kblk = blockIdx.y;            // 0..KCHUNKS-1
    int lane = threadIdx.x;           // wave32
    int l16 = lane & 15;
    int khalf = (lane >> 4) << 1;     // 0 or 2  (K slots {0,1} vs {2,3})

    // A operand: masks rows (M = n). Clamp + zero-scale padded rows so EXEC
    // stays all-ones at the WMMA (required on CDNA5).
    int row = nblk * 16 + l16;
    int rowc = row < NN ? row : NN - 1;
    float validf = row < NN ? 1.0f : 0.0f;
    const float* arow = masks + (size_t)rowc * HWSZ;
    // B operand: B[k=hw][col=t]  ->  Apad row t = lane%16
    const float* brow = Apad + (size_t)l16 * HWSZ;

    const int chunk = HWSZ / KCHUNKS; // 1568
    int hw0 = kblk * chunk;
    int hw1 = hw0 + chunk;

    v8f c = {};
    for (int hw = hw0; hw < hw1; hw += 4) {
        v2f a, b;
        a.x = arow[hw + khalf] * validf;
        a.y = arow[hw + khalf + 1] * validf;
        b.x = brow[hw + khalf];
        b.y = brow[hw + khalf + 1];
        c = __builtin_amdgcn_wmma_f32_16x16x4_f32(
                false, a, false, b, (short)0, c, false, false);
    }

    // D layout: VGPR i -> M = i (lanes 0-15) / 8+i (lanes 16-31), N = lane%16
    int rbase = nblk * 16 + ((lane < 16) ? 0 : 8);
#pragma unroll
    for (int i = 0; i < 8; ++i)
        atomicAdd(&diff[(size_t)(rbase + i) * 16 + l16], c[i]);
}

// PF[t][n] = diff[n,t] - S[t] + class_p[t]  (padded with zeros)
__global__ void finalize_pf_kernel(const float* __restrict__ diff,
                                   const float* __restrict__ S,
                                   const float* __restrict__ cp,
                                   float* __restrict__ PF) {
    int idx = blockIdx.x * blockDim.x + threadIdx.x;
    if (idx >= 16 * NPAD) return;
    int t = idx >> 9;
    int n = idx & (NPAD - 1);
    float v = 0.0f;
    if (t < TT && n < NN) v = diff[(size_t)n * 16 + t] - S[t] + cp[t];
    PF[idx] = v;
}

// GEMM2 + fused epilogue:
//   G[t,hw] = sum_n PF[t,n] * masks[n,hw]        (K = 500, %4 == 0)
//   sal[t,hw]  = G[t,hw] / dcol[hw] / N
//   sal2[hw]   = sum_t cp[t] * sal[t,hw]          (wave32 shfl_xor reduce)
__global__ void __launch_bounds__(256)
gemm_sal_kernel(const float* __restrict__ masks,
                const float* __restrict__ PF,
                const float* __restrict__ dcol,
                const float* __restrict__ cp,
                float* __restrict__ out_sal,
                float* __restrict__ out_sal2) {
    __shared__ float spf[16 * LDPAD];
    __shared__ float scp[16];
    for (int i = threadIdx.x; i < 16 * NPAD; i += 256) {
        int t = i >> 9;
        int n = i & (NPAD - 1);
        spf[t * LDPAD + n] = PF[i];
    }
    if (threadIdx.x < 16)
        scp[threadIdx.x] = (threadIdx.x < TT) ? cp[threadIdx.x] : 0.0f;
    __syncthreads();

    int wave = threadIdx.x >> 5;
    int lane = threadIdx.x & 31;
    int l16 = lane & 15;
    int khalf = (lane >> 4) << 1;

    int tile = blockIdx.x * 8 + wave;       // 0..3135
    int hwbase = tile * 16;

    const float* arow = spf + l16 * LDPAD;  // PF row t = lane%16 (from LDS)
    v8f c = {};
    for (int n0 = 0; n0 < NN; n0 += 4) {
        v2f a, b;
        a.x = arow[n0 + khalf];
        a.y = arow[n0 + khalf + 1];
        b.x = masks[(size_t)(n0 + khalf) * HWSZ + hwbase + l16];
        b.y = masks[(size_t)(n0 + khalf + 1) * HWSZ + hwbase + l16];
        c = __builtin_amdgcn_wmma_f32_16x16x4_f32(
                false, a, false, b, (short)0, c, false, false);
    }

    int hw = hwbase + l16;
    float scale = (1.0f / dcol[hw]) * (1.0f / (float)NN);
    int tbase = (lane < 16) ? 0 : 8;
    float part2 = 0.0f;
#pragma unroll
    for (int i = 0; i < 8; ++i) {
        int t = tbase + i;
        float salv = c[i] * scale;
        if (t < TT) out_sal[(size_t)t * HWSZ + hw] = salv;
        part2 += scp[t] * salv;             // scp[t]=0 for t>=TT
    }
    part2 += __shfl_xor(part2, 16, 32);     // combine t=0..7 with t=8..15
    if (lane < 16) out_sal2[hw] = part2;
}

extern "C" void kernel_launch(void* const* d_in, const int* in_sizes, int n_in,
                              void* d_out, int out_size, void* d_ws, size_t ws_size,
                              hipStream_t stream) {
    const float* x0    = (const float*)d_in[0];   // (1,3,H,W)
    const float* masks = (const float*)d_in[1];   // (N,H,W)
    const float* cp    = (const float*)d_in[2];   // (T,)
    // d_in[3] = random_whole : provably unused (noise cancels)
    const float* Wm    = (const float*)d_in[4];   // (3*H*W, C)
    const int*   tc    = (const int*)d_in[5];     // (T,) target columns

    float* ws   = (float*)d_ws;
    float* Apad = ws + OFF_APAD;
    float* dcol = ws + OFF_DCOL;
    float* S    = ws + OFF_S;
    float* diff = ws + OFF_DIFF;
    float* PF   = ws + OFF_PF;

    float* out_sal  = (float*)d_out;              // (T,H,W)
    float* out_sal2 = out_sal + (size_t)TT * HWSZ; // (H,W)

    init_zero_kernel<<<(NPAD * 16 + 255) / 256, 256, 0, stream>>>(diff, S);
    build_A_kernel<<<HWSZ / 256, 256, 0, stream>>>(x0, Wm, tc, Apad, S);
    colsum_kernel<<<HWSZ / 256, 256, 0, stream>>>(masks, dcol);
    gemm_diff_kernel<<<dim3(NPAD / 16, KCHUNKS), 32, 0, stream>>>(masks, Apad, diff);
    finalize_pf_kernel<<<(16 * NPAD) / 256, 256, 0, stream>>>(diff, S, cp, PF);
    gemm_sal_kernel<<<(HWSZ / 16) / 8, 256, 0, stream>>>(masks, PF, dcol, cp,
                                                         out_sal, out_sal2);
}